// GNNForProtein_46188078301523
// MI455X (gfx1250) — compile-verified
//
#include <hip/hip_runtime.h>
#include <hip/hip_bf16.h>

#define DD 128
#define RT 4              // row tiles per GEMM block (64 rows)

typedef __attribute__((ext_vector_type(2))) float v2f;
typedef __attribute__((ext_vector_type(8))) float v8f;

__device__ __forceinline__ int imin(int a, int b) { return a < b ? a : b; }

// ---------------------------------------------------------------------------
// Eroot[v][j] = sum_k emb[v][k] * Wroot[j][k] + brel[j]     (V x 128, tiny)
// ---------------------------------------------------------------------------
__global__ void eroot_kernel(const float* __restrict__ emb,
                             const float* __restrict__ Wroot,
                             const float* __restrict__ brel,
                             float* __restrict__ Eroot, int V) {
    int idx = blockIdx.x * blockDim.x + threadIdx.x;
    if (idx >= V * DD) return;
    int v = idx >> 7, j = idx & (DD - 1);
    float s = brel[j];
    const float* ev = emb + v * DD;
    const float* wr = Wroot + j * DD;
#pragma unroll 8
    for (int k = 0; k < DD; ++k) s = fmaf(ev[k], wr[k], s);
    Eroot[idx] = s;
}

// ---------------------------------------------------------------------------
// Layer-1 scatter: agg[tgt[e]] += emb[x[src[e]]]   (emb staged in LDS)
// one wave per edge, 4 channels per lane
// ---------------------------------------------------------------------------
__global__ void scatter_emb_kernel(const int* __restrict__ x,
                                   const int* __restrict__ src,
                                   const int* __restrict__ tgt,
                                   const float* __restrict__ emb,
                                   float* __restrict__ agg, int V, int E) {
    extern __shared__ float semb[];
    for (int i = threadIdx.x; i < V * DD; i += blockDim.x) semb[i] = emb[i];
    __syncthreads();
    int e = blockIdx.x * 8 + (threadIdx.x >> 5);
    int lane = threadIdx.x & 31;
    if (e >= E) return;
    int s = src[e];
    int tg = tgt[e];
    int v = x[s];
    const float* sv = semb + v * DD;
    float* ag = agg + (size_t)tg * DD;
#pragma unroll
    for (int q = 0; q < 4; ++q) {
        int c = lane + q * 32;
        atomicAdd(ag + c, sv[c]);
    }
}

// ---------------------------------------------------------------------------
// Layer-2 scatter: agg[tgt[e]] += h1[src[e]]
// ---------------------------------------------------------------------------
__global__ void scatter_feat_kernel(const int* __restrict__ src,
                                    const int* __restrict__ tgt,
                                    const float* __restrict__ h1,
                                    float* __restrict__ agg, int E) {
    int e = blockIdx.x * 8 + (threadIdx.x >> 5);
    int lane = threadIdx.x & 31;
    if (e >= E) return;
    int s = src[e];
    int tg = tgt[e];
    const float* hv = h1 + (size_t)s * DD;
    float* ag = agg + (size_t)tg * DD;
#pragma unroll
    for (int q = 0; q < 4; ++q) {
        int c = lane + q * 32;
        atomicAdd(ag + c, hv[c]);
    }
}

// ---------------------------------------------------------------------------
// Layer-1 GEMM: out = A @ Wrel^T + Eroot[x[row]]
// block = 256 thr = 8 waves; RT=4 16-row tiles / block; wave w -> cols [16w,16w+16)
// B-fragments loaded once per block, reused across all 4 row tiles.
// ---------------------------------------------------------------------------
__global__ void gemm_rel_root_lookup(const float* __restrict__ A,
                                     const float* __restrict__ Wrel,
                                     const float* __restrict__ Eroot,
                                     const int* __restrict__ x,
                                     float* __restrict__ out, int n) {
    __shared__ float sA[RT * 16 * DD];                 // 64x128 f32 = 32KB
    const int row0 = blockIdx.x * (RT * 16);
    const int t = threadIdx.x;
    const int wave = t >> 5, lane = t & 31;

    // cooperative A-tile load (rows clamped)
    for (int i = t; i < (RT * 16 * DD) / 4; i += 256) {
        int e = i * 4;
        int r = e >> 7, c = e & (DD - 1);
        int row = imin(row0 + r, n - 1);
        *(float4*)&sA[e] = *(const float4*)&A[(size_t)row * DD + c];
    }
    __syncthreads();

    const int j0 = wave * 16;
    const int nn = lane & 15;          // N (cols) / M (A rows) selector
    const int mhalf = lane >> 4;       // which K-pair / M-half
    const int kp = mhalf * 2;

    // preload all 32 B fragments once: B[k][n] = Wrel[j0+n][k]
    v2f bfrag[32];
    const float* wp = Wrel + (size_t)(j0 + nn) * DD + kp;
#pragma unroll
    for (int kk = 0; kk < 32; ++kk) bfrag[kk] = *(const v2f*)(wp + kk * 4);

    for (int tt = 0; tt < RT; ++tt) {
        const int trow0 = row0 + tt * 16;

        // C init = Eroot[x[row]][j0+nn]   (root-term via vocabulary lookup)
        v8f acc;
#pragma unroll
        for (int r = 0; r < 8; ++r) {
            int row = imin(trow0 + r + 8 * mhalf, n - 1);
            acc[r] = Eroot[(size_t)x[row] * DD + j0 + nn];
        }

        const float* ap = sA + (tt * 16 + nn) * DD + kp;
#pragma unroll
        for (int kk = 0; kk < 32; ++kk) {
            v2f a = *(const v2f*)(ap + kk * 4);
            acc = __builtin_amdgcn_wmma_f32_16x16x4_f32(
                false, a, false, bfrag[kk], (short)0, acc, false, false);
        }

#pragma unroll
        for (int r = 0; r < 8; ++r) {
            int row = trow0 + r + 8 * mhalf;
            if (row < n) out[(size_t)row * DD + j0 + nn] = acc[r];
        }
    }
}

// ---------------------------------------------------------------------------
// Layer-2 fused GEMM: out = Aagg @ Wrel^T + H1 @ Wroot^T + brel
// ---------------------------------------------------------------------------
__global__ void gemm_fused2(const float* __restrict__ Aagg,
                            const float* __restrict__ H1,
                            const float* __restrict__ Wrel,
                            const float* __restrict__ Wroot,
                            const float* __restrict__ brel,
                            float* __restrict__ out, int n) {
    __shared__ float sA[RT * 16 * DD];                 // 32KB
    __shared__ float sH[RT * 16 * DD];                 // 32KB
    const int row0 = blockIdx.x * (RT * 16);
    const int t = threadIdx.x;
    const int wave = t >> 5, lane = t & 31;

    for (int i = t; i < (RT * 16 * DD) / 4; i += 256) {
        int e = i * 4;
        int r = e >> 7, c = e & (DD - 1);
        int row = imin(row0 + r, n - 1);
        *(float4*)&sA[e] = *(const float4*)&Aagg[(size_t)row * DD + c];
        *(float4*)&sH[e] = *(const float4*)&H1[(size_t)row * DD + c];
    }
    __syncthreads();

    const int j0 = wave * 16;
    const int nn = lane & 15;
    const int mhalf = lane >> 4;
    const int kp = mhalf * 2;

    // preload both B operand sets once per block
    v2f bR[32], bO[32];
    const float* wr = Wrel + (size_t)(j0 + nn) * DD + kp;
    const float* wo = Wroot + (size_t)(j0 + nn) * DD + kp;
#pragma unroll
    for (int kk = 0; kk < 32; ++kk) {
        bR[kk] = *(const v2f*)(wr + kk * 4);
        bO[kk] = *(const v2f*)(wo + kk * 4);
    }

    const float bv = brel[j0 + nn];

    for (int tt = 0; tt < RT; ++tt) {
        const int trow0 = row0 + tt * 16;

        v8f acc;
#pragma unroll
        for (int r = 0; r < 8; ++r) acc[r] = bv;

        const float* ap = sA + (tt * 16 + nn) * DD + kp;
        const float* hp = sH + (tt * 16 + nn) * DD + kp;
#pragma unroll
        for (int kk = 0; kk < 32; ++kk) {
            v2f a = *(const v2f*)(ap + kk * 4);
            acc = __builtin_amdgcn_wmma_f32_16x16x4_f32(
                false, a, false, bR[kk], (short)0, acc, false, false);
        }
#pragma unroll
        for (int kk = 0; kk < 32; ++kk) {
            v2f a = *(const v2f*)(hp + kk * 4);
            acc = __builtin_amdgcn_wmma_f32_16x16x4_f32(
                false, a, false, bO[kk], (short)0, acc, false, false);
        }

#pragma unroll
        for (int r = 0; r < 8; ++r) {
            int row = trow0 + r + 8 * mhalf;
            if (row < n) out[(size_t)row * DD + j0 + nn] = acc[r];
        }
    }
}

// ---------------------------------------------------------------------------
// Segment-mean pooling: one block per protein, thread c sums channel c
// ---------------------------------------------------------------------------
__global__ void pool_kernel(const float* __restrict__ h2,
                            const int* __restrict__ ends,
                            float* __restrict__ out, int n) {
    int p = blockIdx.x;
    int c = threadIdx.x;  // 128 threads
    int start = (p == 0) ? 0 : ends[p - 1];
    int end = ends[p];
    if (end > n) end = n;
    float s = 0.f;
    for (int i = start; i < end; ++i) s += h2[(size_t)i * DD + c];
    int cnt = end - start;
    out[(size_t)p * DD + c] = (cnt > 0) ? s / (float)cnt : 0.f;
}

// ---------------------------------------------------------------------------
extern "C" void kernel_launch(void* const* d_in, const int* in_sizes, int n_in,
                              void* d_out, int out_size, void* d_ws, size_t ws_size,
                              hipStream_t stream) {
    const int* x      = (const int*)d_in[0];
    const int* ei     = (const int*)d_in[1];
    const int* ends   = (const int*)d_in[2];
    const float* emb   = (const float*)d_in[3];
    const float* Wrel1 = (const float*)d_in[4];
    const float* brel1 = (const float*)d_in[5];
    const float* Wroot1= (const float*)d_in[6];
    const float* Wrel2 = (const float*)d_in[7];
    const float* brel2 = (const float*)d_in[8];
    const float* Wroot2= (const float*)d_in[9];
    float* out = (float*)d_out;

    const int n  = in_sizes[0];
    const int E  = in_sizes[1] / 2;
    const int Bp = in_sizes[2];
    const int V  = in_sizes[3] / DD;
    const int* src = ei;
    const int* tgt = ei + E;

    const size_t nf = (size_t)n * DD;
    float* bufA  = (float*)d_ws;       // agg1, later reused as agg2
    float* h1    = bufA + nf;
    float* h2    = h1 + nf;
    float* Eroot = h2 + nf;            // V*128 floats

    const int tiles   = (n + RT * 16 - 1) / (RT * 16);
    const int eblocks = (E + 7) / 8;

    // --- layer 1 ---
    hipMemsetAsync(bufA, 0, nf * sizeof(float), stream);
    eroot_kernel<<<(V * DD + 255) / 256, 256, 0, stream>>>(emb, Wroot1, brel1, Eroot, V);
    scatter_emb_kernel<<<eblocks, 256, (size_t)V * DD * sizeof(float), stream>>>(
        x, src, tgt, emb, bufA, V, E);
    gemm_rel_root_lookup<<<tiles, 256, 0, stream>>>(bufA, Wrel1, Eroot, x, h1, n);

    // --- layer 2 ---
    hipMemsetAsync(bufA, 0, nf * sizeof(float), stream);
    scatter_feat_kernel<<<eblocks, 256, 0, stream>>>(src, tgt, h1, bufA, E);
    gemm_fused2<<<tiles, 256, 0, stream>>>(bufA, h1, Wrel2, Wroot2, brel2, h2, n);

    // --- pooling ---
    pool_kernel<<<Bp, DD, 0, stream>>>(h2, ends, out, n);
}